// Scalar_90494960927294
// MI455X (gfx1250) — compile-verified
//
#include <hip/hip_runtime.h>
#include <stdint.h>

#define NYg 444
#define NXg 444
#define NYX (NYg*NXg)
#define NSs 2
#define NSRC 1
#define NREC 100
#define NTT 100
#define PADc 22
#define DTc 0.0005f

// LDS tile: rows ty0-2..ty0+17, cols tx0-2..tx0+17 (20x20 used, stride 24)
#define TH 20
#define TW 20
#define TS 24

#define AS1 __attribute__((address_space(1)))
#define AS3 __attribute__((address_space(3)))

#define HAVE_ASYNC_LDS __has_builtin(__builtin_amdgcn_global_load_async_to_lds_b32)

typedef __attribute__((ext_vector_type(2))) float v2f;
typedef __attribute__((ext_vector_type(8))) float v8f;

__device__ __forceinline__ v8f wmma4(v2f a, v2f b, v8f c) {
  // V_WMMA_F32_16X16X4_F32 : D(16x16,f32) = A(16x4,f32) * B(4x16,f32) + C
  return __builtin_amdgcn_wmma_f32_16x16x4_f32(false, a, false, b, (short)0, c, false, false);
}

// branchless zero-padded load: clamp address (always legal), select 0 if OOB
__device__ __forceinline__ float ldc(const float* __restrict__ F, int y, int x) {
  int yc = min(max(y, 0), NYg - 1);
  int xc = min(max(x, 0), NXg - 1);
  float val = F[yc * NXg + xc];
  bool inb = ((unsigned)y < (unsigned)NYg) && ((unsigned)x < (unsigned)NXg);
  return inb ? val : 0.0f;
}

// first-derivative band coefficients (offset -2..+2) / (12*h), h=4
#define C1_0 ( 1.0f/48.0f)
#define C1_1 (-8.0f/48.0f)
#define C1_2 ( 0.0f)
#define C1_3 ( 8.0f/48.0f)
#define C1_4 (-1.0f/48.0f)
// second-derivative band coefficients / (12*h*h)
#define C2_0 ( -1.0f/192.0f)
#define C2_1 ( 16.0f/192.0f)
#define C2_2 (-30.0f/192.0f)
#define C2_3 ( 16.0f/192.0f)
#define C2_4 ( -1.0f/192.0f)

__device__ __forceinline__ float c1val(int i) {
  float r = 0.0f;
  r = (i == 0) ? C1_0 : r;
  r = (i == 1) ? C1_1 : r;
  r = (i == 3) ? C1_3 : r;
  r = (i == 4) ? C1_4 : r;
  return r;
}
__device__ __forceinline__ float c2val(int i) {
  float r = 0.0f;
  r = (i == 0) ? C2_0 : r;
  r = (i == 1) ? C2_1 : r;
  r = (i == 2) ? C2_2 : r;
  r = (i == 3) ? C2_3 : r;
  r = (i == 4) ? C2_4 : r;
  return r;
}

// banded-matrix coefficient gather from an 8-entry LDS table (slots 5..7 are zero)
__device__ __forceinline__ float band(const float* tab, int d) {
  int idx = ((unsigned)d <= 4u) ? d : 5;
  return tab[idx];
}

// stage wfc halo tile (20x20) into LDS; async fast path for interior tiles
__device__ __forceinline__ void stage_tile(float* tile, const float* __restrict__ W,
                                           int ty0, int tx0, int lane) {
  const int y0 = ty0 - 2, x0 = tx0 - 2;
  const bool interior = (y0 >= 0) && (y0 + TH <= NYg) && (x0 >= 0) && (x0 + TW <= NXg);
  if (interior) {
#if HAVE_ASYNC_LDS
    for (int i = lane; i < TH * TW; i += 32) {
      int r = i / TW, c = i - r * TW;
      const float* gp = W + (y0 + r) * NXg + (x0 + c);
      float* lp = tile + r * TS + c;
      __builtin_amdgcn_global_load_async_to_lds_b32(
          (AS1 int*)(uintptr_t)gp, (AS3 int*)(uintptr_t)lp, 0, 0);
    }
#if __has_builtin(__builtin_amdgcn_s_wait_asynccnt)
    __builtin_amdgcn_s_wait_asynccnt(0);
#else
    asm volatile("s_wait_asynccnt 0x0" ::: "memory");
#endif
#else
    for (int i = lane; i < TH * TW; i += 32) {
      int r = i / TW, c = i - r * TW;
      tile[r * TS + c] = W[(y0 + r) * NXg + (x0 + c)];
    }
#endif
  } else {
    for (int i = lane; i < TH * TW; i += 32) {
      int r = i / TW, c = i - r * TW;
      tile[r * TS + c] = ldc(W, y0 + r, x0 + c);
    }
  }
  __syncthreads();   // single-wave workgroup: barrier op is ~free, forces LDS visibility
}

// ---------------- init kernels ----------------
__global__ void k_zero(float* p, int n) {
  int i = blockIdx.x * blockDim.x + threadIdx.x;
  if (i < n) p[i] = 0.0f;
}

__global__ void k_model(const float* __restrict__ v, float* __restrict__ v2dt2,
                        float* __restrict__ bprof) {
  int i = blockIdx.x * blockDim.x + threadIdx.x;
  if (i < NYX) {
    int y = i / NXg, x = i % NXg;
    int vy = min(max(y - PADc, 0), 399);
    int vx = min(max(x - PADc, 0), 399);
    float vv = v[vy * 400 + vx];
    v2dt2[i] = vv * vv * (DTc * DTc);
  }
  if (i < NYg) {
    float fi = (float)i;
    float dlo = fminf(fmaxf((22.0f - fi) / 20.0f, 0.0f), 1.0f);
    float dhi = fminf(fmaxf((fi - 421.0f) / 20.0f, 0.0f), 1.0f);
    float frac = fmaxf(dlo, dhi);
    const float sigma_max = 3.0f * 3000.0f * 6.907755279f / (2.0f * 20.0f * 4.0f);
    float sigma = sigma_max * frac * frac;
    bprof[i] = expf(-sigma * DTc);
  }
}

// ---------------- CPML psi update: psi = b*psi + (b-1)*d1(wfc) ----------------
__global__ __launch_bounds__(32) void k_psi(const float* __restrict__ wfc,
                                            float* __restrict__ psiy,
                                            float* __restrict__ psix,
                                            const float* __restrict__ bprof) {
  __shared__ float tile[TH * TS];
  __shared__ float c1tab[8];
  const int s   = blockIdx.z;
  const int ty0 = blockIdx.y * 16;
  const int tx0 = blockIdx.x * 16;
  const int lane = threadIdx.x;
  const int hi = lane >> 4;      // lanes 16..31
  const int mn = lane & 15;      // A: row M ; B/D: column N
  const float* __restrict__ W = wfc + s * NYX;

  if (lane < 8) c1tab[lane] = c1val(lane);
  stage_tile(tile, W, ty0, tx0, lane);

  // d1y(wfc) over a 16x16 tile: O = C1(16x20) * U(20x16) via 5x v_wmma_f32_16x16x4_f32
  v8f acc1 = {0.f,0.f,0.f,0.f,0.f,0.f,0.f,0.f};
#pragma unroll
  for (int kc = 0; kc < 5; ++kc) {
    int k0 = 4 * kc + (hi ? 2 : 0);
    int k1 = k0 + 1;
    v2f a, b;
    a.x = band(c1tab, k0 - mn);
    a.y = band(c1tab, k1 - mn);
    b.x = tile[k0 * TS + mn + 2];
    b.y = tile[k1 * TS + mn + 2];
    acc1 = wmma4(a, b, acc1);
  }

#pragma unroll
  for (int r = 0; r < 8; ++r) {
    int m = r + 8 * hi;          // D layout: vgpr r -> row r (lanes 0-15) / r+8 (lanes 16-31)
    int y = ty0 + m;
    int x = tx0 + mn;
    if (y < NYg && x < NXg) {
      int p = s * NYX + y * NXg + x;
      float byv = bprof[y];
      psiy[p] = byv * psiy[p] + (byv - 1.0f) * acc1[r];
      float bxv = bprof[x];
      const float* row = &tile[(m + 2) * TS + (mn + 2)];
      float d1x = (8.0f * (row[1] - row[-1]) - (row[2] - row[-2])) * (1.0f / 48.0f);
      psix[p] = bxv * psix[p] + (bxv - 1.0f) * d1x;
    }
  }
}

// ---------------- wave update: zeta, leapfrog; writes wfn into wfp buffer ----------------
__global__ __launch_bounds__(32) void k_wave(const float* __restrict__ wfc,
                                             float* __restrict__ wfp,
                                             const float* __restrict__ psiy,
                                             const float* __restrict__ psix,
                                             float* __restrict__ zetay,
                                             float* __restrict__ zetax,
                                             const float* __restrict__ v2dt2,
                                             const float* __restrict__ bprof) {
  __shared__ float tile[TH * TS];
  __shared__ float c1tab[8];
  __shared__ float c2tab[8];
  const int s   = blockIdx.z;
  const int ty0 = blockIdx.y * 16;
  const int tx0 = blockIdx.x * 16;
  const int lane = threadIdx.x;
  const int hi = lane >> 4;
  const int mn = lane & 15;
  const float* __restrict__ W  = wfc  + s * NYX;
  const float* __restrict__ PY = psiy + s * NYX;
  const float* __restrict__ PX = psix + s * NYX;

  if (lane < 8) { c1tab[lane] = c1val(lane); c2tab[lane] = c2val(lane); }
  stage_tile(tile, W, ty0, tx0, lane);

  // acc2 = d2y(wfc) (from LDS tile), accp = d1y(psiy) (global, branchless clamped)
  v8f acc2 = {0.f,0.f,0.f,0.f,0.f,0.f,0.f,0.f};
  v8f accp = {0.f,0.f,0.f,0.f,0.f,0.f,0.f,0.f};
  const int yb = ty0 - 2;
  const int xg = tx0 + mn;
#pragma unroll
  for (int kc = 0; kc < 5; ++kc) {
    int k0 = 4 * kc + (hi ? 2 : 0);
    int k1 = k0 + 1;
    v2f a2, a1, bw, bp;
    a2.x = band(c2tab, k0 - mn);
    a2.y = band(c2tab, k1 - mn);
    a1.x = band(c1tab, k0 - mn);
    a1.y = band(c1tab, k1 - mn);
    bw.x = tile[k0 * TS + mn + 2];
    bw.y = tile[k1 * TS + mn + 2];
    bp.x = ldc(PY, yb + k0, xg);
    bp.y = ldc(PY, yb + k1, xg);
    acc2 = wmma4(a2, bw, acc2);
    accp = wmma4(a1, bp, accp);
  }

#pragma unroll
  for (int r = 0; r < 8; ++r) {
    int m = r + 8 * hi;
    int y = ty0 + m;
    int x = xg;
    if (y < NYg && x < NXg) {
      int p = s * NYX + y * NXg + x;
      const float* row = &tile[(m + 2) * TS + (mn + 2)];
      float wc = row[0];
      float ly = acc2[r] + accp[r];
      float d2x = (-(row[2] + row[-2]) + 16.0f * (row[1] + row[-1]) - 30.0f * wc)
                  * (1.0f / 192.0f);
      float d1px = (8.0f * (ldc(PX, y, x + 1) - ldc(PX, y, x - 1))
                         - (ldc(PX, y, x + 2) - ldc(PX, y, x - 2))) * (1.0f / 48.0f);
      float lx = d2x + d1px;
      float byv = bprof[y], bxv = bprof[x];
      float zy = byv * zetay[p] + (byv - 1.0f) * ly;
      float zx = bxv * zetax[p] + (bxv - 1.0f) * lx;
      zetay[p] = zy;
      zetax[p] = zx;
      wfp[p] = v2dt2[y * NXg + x] * (ly + zy + lx + zx) + 2.0f * wc - wfp[p];
    }
  }
}

// ---------------- source injection (after wave update, before recording) ----------------
__global__ void k_src(float* __restrict__ wfn, const float* __restrict__ v2dt2,
                      const float* __restrict__ amp, const long long* __restrict__ slocs,
                      int t) {
  int i = threadIdx.x;
  if (i < NSs * NSRC) {
    int s = i / NSRC, j = i % NSRC;
    int sy = (int)slocs[(s * NSRC + j) * 2 + 0] + PADc;
    int sx = (int)slocs[(s * NSRC + j) * 2 + 1] + PADc;
    float a = amp[(s * NSRC + j) * NTT + t];
    wfn[s * NYX + sy * NXg + sx] += v2dt2[sy * NXg + sx] * a;
  }
}

// ---------------- receiver gather ----------------
__global__ void k_rec(const float* __restrict__ wfn, const long long* __restrict__ rlocs,
                      float* __restrict__ rec, int t) {
  int i = blockIdx.x * blockDim.x + threadIdx.x;
  if (i < NSs * NREC) {
    int s = i / NREC, r = i % NREC;
    int ry = (int)rlocs[(s * NREC + r) * 2 + 0] + PADc;
    int rx = (int)rlocs[(s * NREC + r) * 2 + 1] + PADc;
    rec[(s * NREC + r) * NTT + t] = wfn[s * NYX + ry * NXg + rx];
  }
}

extern "C" void kernel_launch(void* const* d_in, const int* in_sizes, int n_in,
                              void* d_out, int out_size, void* d_ws, size_t ws_size,
                              hipStream_t stream) {
  (void)in_sizes; (void)n_in; (void)out_size; (void)ws_size;
  const float*     v_in  = (const float*)d_in[0];
  const float*     amp   = (const float*)d_in[1];
  const long long* slocs = (const long long*)d_in[2];
  const long long* rlocs = (const long long*)d_in[3];

  float* ws    = (float*)d_ws;
  float* wfA   = ws;
  float* wfB   = wfA   + NSs * NYX;
  float* psiy  = wfB   + NSs * NYX;
  float* psix  = psiy  + NSs * NYX;
  float* zetay = psix  + NSs * NYX;
  float* zetax = zetay + NSs * NYX;
  float* v2dt2 = zetax + NSs * NYX;
  float* bprof = v2dt2 + NYX;

  float* out_f   = (float*)d_out;
  float* rec_out = out_f + 6 * NSs * NYX;

  int nz = 6 * NSs * NYX;
  k_zero<<<(nz + 255) / 256, 256, 0, stream>>>(ws, nz);
  k_model<<<(NYX + 255) / 256, 256, 0, stream>>>(v_in, v2dt2, bprof);

  dim3 grid((NXg + 15) / 16, (NYg + 15) / 16, NSs);  // 28 x 28 x 2, one wave32 per 16x16 tile
  float* cur = wfA;   // wfc
  float* prev = wfB;  // wfp ; wfn is written into prev, then swapped

  for (int t = 0; t < NTT; ++t) {
    k_psi<<<grid, 32, 0, stream>>>(cur, psiy, psix, bprof);
    k_wave<<<grid, 32, 0, stream>>>(cur, prev, psiy, psix, zetay, zetax, v2dt2, bprof);
    k_src<<<1, 32, 0, stream>>>(prev, v2dt2, amp, slocs, t);
    k_rec<<<1, 256, 0, stream>>>(prev, rlocs, rec_out, t);
    float* tmp = cur; cur = prev; prev = tmp;
  }

  size_t bytes = (size_t)NSs * NYX * sizeof(float);
  (void)hipMemcpyAsync(out_f + 0 * NSs * NYX, cur,   bytes, hipMemcpyDeviceToDevice, stream); // wfc
  (void)hipMemcpyAsync(out_f + 1 * NSs * NYX, prev,  bytes, hipMemcpyDeviceToDevice, stream); // wfp
  (void)hipMemcpyAsync(out_f + 2 * NSs * NYX, psiy,  bytes, hipMemcpyDeviceToDevice, stream);
  (void)hipMemcpyAsync(out_f + 3 * NSs * NYX, psix,  bytes, hipMemcpyDeviceToDevice, stream);
  (void)hipMemcpyAsync(out_f + 4 * NSs * NYX, zetay, bytes, hipMemcpyDeviceToDevice, stream);
  (void)hipMemcpyAsync(out_f + 5 * NSs * NYX, zetax, bytes, hipMemcpyDeviceToDevice, stream);
}